// GNNModel_69423851373024
// MI455X (gfx1250) — compile-verified
//
#include <hip/hip_runtime.h>

#define N_NODES  100000
#define N_EDGES  3200000
#define N_GRAPHS 1024
#define IN_CH    6
#define HID      128
#define OUT_DIM  90

typedef __attribute__((ext_vector_type(2))) float v2f;
typedef __attribute__((ext_vector_type(8))) float v8f;

__device__ __forceinline__ void fatomic_add(float* p, float v) {
    (void)unsafeAtomicAdd(p, v);   // lowers to global_atomic_add_f32 on gfx1250
}

// ---------------- degree / normalization ----------------
__global__ void k_init_deg(float* __restrict__ deg) {
    int i = blockIdx.x * blockDim.x + threadIdx.x;
    if (i < N_NODES) deg[i] = 1.0f;             // self-loop contribution
}

__global__ void k_deg_accum(const int* __restrict__ dst, float* __restrict__ deg) {
    int e = blockIdx.x * blockDim.x + threadIdx.x;
    if (e < N_EDGES) fatomic_add(&deg[dst[e]], 1.0f);
}

__global__ void k_dinv(const float* __restrict__ deg, float* __restrict__ dinv) {
    int i = blockIdx.x * blockDim.x + threadIdx.x;
    if (i < N_NODES) {
        float d = deg[i];
        dinv[i] = (d > 0.0f) ? rsqrtf(d) : 0.0f;
    }
}

// ---------------- layer-1 GEMM (K=6, scalar) ----------------
__global__ void k_gemm1(const float* __restrict__ x, const float* __restrict__ W1,
                        float* __restrict__ h) {
    int idx = blockIdx.x * blockDim.x + threadIdx.x;
    if (idx >= N_NODES * HID) return;
    int row = idx >> 7, col = idx & (HID - 1);
    const float* xr = x + row * IN_CH;
    float acc = 0.0f;
#pragma unroll
    for (int k = 0; k < IN_CH; ++k) acc += xr[k] * W1[k * HID + col];
    h[idx] = acc;
}

// ---------------- GCN aggregation ----------------
__global__ void k_selfloop(const float* __restrict__ h, const float* __restrict__ dinv,
                           float* __restrict__ agg) {
    int idx = blockIdx.x * blockDim.x + threadIdx.x;
    if (idx >= N_NODES * HID) return;
    int n = idx >> 7;
    float di = dinv[n];
    agg[idx] = h[idx] * di * di;
}

__global__ void k_edge_scatter(const float* __restrict__ h, const int* __restrict__ src,
                               const int* __restrict__ dst, const float* __restrict__ dinv,
                               float* __restrict__ agg) {
    long long tid = (long long)blockIdx.x * blockDim.x + threadIdx.x;
    int e    = (int)(tid >> 5);
    int lane = (int)(tid & 31);
    if (e >= N_EDGES) return;
    int s = src[e], d = dst[e];
    float norm = dinv[s] * dinv[d];
    const float4 v = *reinterpret_cast<const float4*>(h + (size_t)s * HID + lane * 4);
    float* base = agg + (size_t)d * HID + lane * 4;
    fatomic_add(base + 0, v.x * norm);
    fatomic_add(base + 1, v.y * norm);
    fatomic_add(base + 2, v.z * norm);
    fatomic_add(base + 3, v.w * norm);
}

__global__ void k_bias_relu(float* __restrict__ a, const float* __restrict__ b) {
    int idx = blockIdx.x * blockDim.x + threadIdx.x;
    if (idx >= N_NODES * HID) return;
    float v = a[idx] + b[idx & (HID - 1)];
    a[idx] = v > 0.0f ? v : 0.0f;
}

// ---------------- layer-2 GEMM: fp32 WMMA 16x16x4 ----------------
// One block = 8 waves = one 16-row stripe of A; wave w owns N-tile w.
// A tile staged in LDS with padded row stride (132) to avoid bank conflicts.
__global__ void __launch_bounds__(256)
k_gemm2_wmma(const float* __restrict__ A, const float* __restrict__ B,
             float* __restrict__ C) {
    __shared__ float As[16 * 132];
    const int m0 = blockIdx.x * 16;
    const int t  = threadIdx.x;

#pragma unroll
    for (int i = 0; i < 8; ++i) {
        int idx = t + i * 256;              // 0..2047
        int r = idx >> 7, k = idx & 127;
        As[r * 132 + k] = A[(size_t)m0 * HID + idx];
    }
    __syncthreads();

    const int wave = t >> 5;                // N-tile index 0..7
    const int lane = t & 31;
    const int col  = lane & 15;
    const int row  = lane & 15;             // A rows M=0..15 in both half-waves
    const int koff = (lane >> 4) * 2;       // lanes 16-31 hold K+2,K+3
    const int n0   = wave * 16;

    v8f acc = {};
#pragma unroll 4
    for (int k0 = 0; k0 < HID; k0 += 4) {
        v2f a, b;
        a.x = As[row * 132 + k0 + koff];
        a.y = As[row * 132 + k0 + koff + 1];
        b.x = B[(k0 + koff) * HID + n0 + col];
        b.y = B[(k0 + koff + 1) * HID + n0 + col];
        acc = __builtin_amdgcn_wmma_f32_16x16x4_f32(
            /*neg_a=*/false, a, /*neg_b=*/false, b,
            /*c_mod=*/(short)0, acc, /*reuse_a=*/false, /*reuse_b=*/false);
    }

    const int rowbase = (lane >> 4) * 8;    // VGPR i -> rows i / i+8
#pragma unroll
    for (int i = 0; i < 8; ++i)
        C[(size_t)(m0 + rowbase + i) * HID + n0 + col] = acc[i];
}

// ---------------- global mean pool + FC ----------------
__global__ void k_pool_zero(float* __restrict__ sums, float* __restrict__ cnt) {
    int idx = blockIdx.x * blockDim.x + threadIdx.x;
    if (idx < N_GRAPHS * HID) sums[idx] = 0.0f;
    if (idx < N_GRAPHS) cnt[idx] = 0.0f;
}

__global__ void k_pool(const float* __restrict__ act, const int* __restrict__ batch,
                       float* __restrict__ sums, float* __restrict__ cnt) {
    long long tid = (long long)blockIdx.x * blockDim.x + threadIdx.x;
    int n    = (int)(tid >> 5);
    int lane = (int)(tid & 31);
    if (n >= N_NODES) return;
    int g = batch[n];
    const float4 v = *reinterpret_cast<const float4*>(act + (size_t)n * HID + lane * 4);
    float* base = sums + (size_t)g * HID + lane * 4;
    fatomic_add(base + 0, v.x);
    fatomic_add(base + 1, v.y);
    fatomic_add(base + 2, v.z);
    fatomic_add(base + 3, v.w);
    if (lane == 0) fatomic_add(&cnt[g], 1.0f);
}

__global__ void k_fc(const float* __restrict__ sums, const float* __restrict__ cnt,
                     const float* __restrict__ Wfc, const float* __restrict__ bfc,
                     float* __restrict__ out) {
    __shared__ float p[HID];
    int g = blockIdx.x, t = threadIdx.x;
    float inv = 1.0f / fmaxf(cnt[g], 1.0f);
    p[t] = sums[g * HID + t] * inv;
    __syncthreads();
    if (t < OUT_DIM) {
        float acc = bfc[t];
#pragma unroll 8
        for (int k = 0; k < HID; ++k) acc += p[k] * Wfc[k * OUT_DIM + t];
        out[g * OUT_DIM + t] = acc;
    }
}

// ---------------- launch ----------------
extern "C" void kernel_launch(void* const* d_in, const int* in_sizes, int n_in,
                              void* d_out, int out_size, void* d_ws, size_t ws_size,
                              hipStream_t stream) {
    const float* x    = (const float*)d_in[0];
    const int*   ei   = (const int*)d_in[1];      // [2, E]
    const int*   bat  = (const int*)d_in[2];
    const float* W1   = (const float*)d_in[3];
    const float* b1   = (const float*)d_in[4];
    const float* W2   = (const float*)d_in[5];
    const float* b2   = (const float*)d_in[6];
    const float* Wfc  = (const float*)d_in[7];
    const float* bfc  = (const float*)d_in[8];
    float*       out  = (float*)d_out;

    const int* src = ei;
    const int* dst = ei + N_EDGES;

    const size_t NH = (size_t)N_NODES * HID;      // 12.8M floats
    float* ws   = (float*)d_ws;
    float* bufA = ws;                             // h (pre-aggregation)
    float* bufB = ws + NH;                        // agg / activations
    float* deg  = ws + 2 * NH;
    float* dinv = deg + N_NODES;
    float* sums = dinv + N_NODES;
    float* cntg = sums + (size_t)N_GRAPHS * HID;

    const int TB = 256;
    const int gNode = (N_NODES + TB - 1) / TB;
    const int gEdge = (N_EDGES + TB - 1) / TB;
    const int gNH   = (int)((NH + TB - 1) / TB);
    const int gEdgeW = (int)(((long long)N_EDGES * 32 + TB - 1) / TB);
    const int gNodeW = (int)(((long long)N_NODES * 32 + TB - 1) / TB);

    // normalization
    k_init_deg<<<gNode, TB, 0, stream>>>(deg);
    k_deg_accum<<<gEdge, TB, 0, stream>>>(dst, deg);
    k_dinv<<<gNode, TB, 0, stream>>>(deg, dinv);

    // layer 1
    k_gemm1<<<gNH, TB, 0, stream>>>(x, W1, bufA);
    k_selfloop<<<gNH, TB, 0, stream>>>(bufA, dinv, bufB);
    k_edge_scatter<<<gEdgeW, TB, 0, stream>>>(bufA, src, dst, dinv, bufB);
    k_bias_relu<<<gNH, TB, 0, stream>>>(bufB, b1);

    // layer 2
    k_gemm2_wmma<<<N_NODES / 16, TB, 0, stream>>>(bufB, W2, bufA);
    k_selfloop<<<gNH, TB, 0, stream>>>(bufA, dinv, bufB);
    k_edge_scatter<<<gEdgeW, TB, 0, stream>>>(bufA, src, dst, dinv, bufB);
    k_bias_relu<<<gNH, TB, 0, stream>>>(bufB, b2);

    // pool + fc
    k_pool_zero<<<(N_GRAPHS * HID + TB - 1) / TB, TB, 0, stream>>>(sums, cntg);
    k_pool<<<gNodeW, TB, 0, stream>>>(bufB, bat, sums, cntg);
    k_fc<<<N_GRAPHS, HID, 0, stream>>>(sums, cntg, Wfc, bfc, out);
}